// MultiHeadAttention_39444979646805
// MI455X (gfx1250) — compile-verified
//
#include <hip/hip_runtime.h>
#include <hip/hip_bf16.h>
#include <math.h>

// ---------------------------------------------------------------------------
// L2-distance multi-head attention for MI455X (gfx1250, wave32).
// All matmuls on V_WMMA_F32_16X16X4_F32 (exact fp32 — the
// sqrt(|q|^2+|k|^2-2qk) distance is cancellation-sensitive).
// Attention K/V tiles double-buffered in LDS via GLOBAL_LOAD_ASYNC_TO_LDS
// (ASYNCcnt path), overlapping copies for tile j+1 with WMMAs for tile j.
//
// Shapes: B=8, N=1024, E=768, H=12, D=64.
// ---------------------------------------------------------------------------

#define B_ 8
#define N_ 1024
#define E_ 768
#define H_ 12
#define D_ 64

typedef float v2f __attribute__((ext_vector_type(2)));
typedef float v8f __attribute__((ext_vector_type(8)));

static __device__ __forceinline__ v8f wmma_f32(v2f a, v2f b, v8f c) {
    return __builtin_amdgcn_wmma_f32_16x16x4_f32(
        false, a, false, b, (short)0, c, false, false);
}

// 16-WMMA macro-step for a 64x64 wave tile (4 A-frags x 4 B-frags).
static __device__ __forceinline__ void mma16(v8f acc[4][4], const v2f a[4],
                                             const v2f b[4]) {
#pragma unroll
    for (int m = 0; m < 4; ++m)
#pragma unroll
        for (int t = 0; t < 4; ++t)
            acc[m][t] = wmma_f32(a[m], b[t], acc[m][t]);
}

// Async copy of 8 bytes global -> LDS (per lane). VDST = LDS byte offset,
// VADDR = 64-bit global address. Tracked by ASYNCcnt.
static __device__ __forceinline__ void async_copy_b64(unsigned lds_off,
                                                      const void* gptr) {
    asm volatile("global_load_async_to_lds_b64 %0, %1, off"
                 :: "v"(lds_off), "v"((unsigned long long)(size_t)gptr)
                 : "memory");
}
static __device__ __forceinline__ void wait_async0() {
    asm volatile("s_wait_asynccnt 0x0" ::: "memory");
}
static __device__ __forceinline__ unsigned lds_off(const void* p) {
    return (unsigned)(size_t)p;   // low 32 bits of generic addr == LDS offset
}

// ---------------------------------------------------------------------------
// Kernel 1: qkv = x @ W_qkv^T ; scatter into Q[B,H,N,D], V[B,H,N,D].
// Wave computes a 64(M) x 64(N) tile. K-loop manually unrolled by 2 with
// ping-pong fragment registers (no copy movs, loads overlap 16-WMMA bursts).
// Grid: 128 M-tiles x 24 N-groups = 3072 waves = 768 blocks of 128.
// ---------------------------------------------------------------------------
__global__ void __launch_bounds__(128)
qkv_gemm_kernel(const float* __restrict__ x, const float* __restrict__ Wqkv,
                float* __restrict__ Qbuf, float* __restrict__ Vbuf) {
    const int lane = threadIdx.x & 31;
    const int warp = threadIdx.x >> 5;
    const int wid  = blockIdx.x * 4 + warp;       // 0..3071
    const int mtile = wid % 128;                  // 64-row M tiles
    const int ngrp  = wid / 128;                  // 0..23
    const int lm = lane & 15;
    const int hi = lane >> 4;

    const float* xrow[4];
    const float* wrow[4];
#pragma unroll
    for (int m = 0; m < 4; ++m)
        xrow[m] = x + (size_t)(mtile * 64 + m * 16 + lm) * E_ + 2 * hi;
#pragma unroll
    for (int t = 0; t < 4; ++t)
        wrow[t] = Wqkv + (size_t)(ngrp * 64 + t * 16 + lm) * E_ + 2 * hi;

    v8f acc[4][4] = {};
    v2f a0[4], b0[4], a1[4], b1[4];
#pragma unroll
    for (int m = 0; m < 4; ++m) a0[m] = *(const v2f*)(xrow[m]);
#pragma unroll
    for (int t = 0; t < 4; ++t) b0[t] = *(const v2f*)(wrow[t]);

    for (int k0 = 0; k0 < E_ - 8; k0 += 8) {
#pragma unroll
        for (int m = 0; m < 4; ++m) a1[m] = *(const v2f*)(xrow[m] + k0 + 4);
#pragma unroll
        for (int t = 0; t < 4; ++t) b1[t] = *(const v2f*)(wrow[t] + k0 + 4);
        mma16(acc, a0, b0);
#pragma unroll
        for (int m = 0; m < 4; ++m) a0[m] = *(const v2f*)(xrow[m] + k0 + 8);
#pragma unroll
        for (int t = 0; t < 4; ++t) b0[t] = *(const v2f*)(wrow[t] + k0 + 8);
        mma16(acc, a1, b1);
    }
    {   // k0 = E_-8: last two K-steps
#pragma unroll
        for (int m = 0; m < 4; ++m) a1[m] = *(const v2f*)(xrow[m] + E_ - 4);
#pragma unroll
        for (int t = 0; t < 4; ++t) b1[t] = *(const v2f*)(wrow[t] + E_ - 4);
        mma16(acc, a0, b0);
        mma16(acc, a1, b1);
    }

    // C/D layout: element v of lane L is (M = v + 8*hi, Ncol = L%16).
#pragma unroll
    for (int m = 0; m < 4; ++m) {
#pragma unroll
        for (int t = 0; t < 4; ++t) {
            const int c = ngrp * 64 + t * 16 + lm;    // 0..1535
#pragma unroll
            for (int v = 0; v < 8; ++v) {
                const int r  = mtile * 64 + m * 16 + v + 8 * hi;  // 0..8191
                const int bb = r >> 10, n = r & 1023;
                const float val = acc[m][t][v];
                if (c < E_) {
                    const int h = c >> 6, dd = c & 63;
                    Qbuf[(((size_t)bb * H_ + h) * N_ + n) * D_ + dd] = val;
                } else {
                    const int c2 = c - E_;
                    const int h = c2 >> 6, dd = c2 & 63;
                    Vbuf[(((size_t)bb * H_ + h) * N_ + n) * D_ + dd] = val;
                }
            }
        }
    }
}

// ---------------------------------------------------------------------------
// Kernel 2: sq[b,h,n] = sum_d Q^2. One thread per row (98304 rows).
// ---------------------------------------------------------------------------
__global__ void sqnorm_kernel(const float* __restrict__ Qbuf,
                              float* __restrict__ sq) {
    const int r = blockIdx.x * blockDim.x + threadIdx.x;   // < 98304
    const float* p = Qbuf + (size_t)r * D_;
    float s = 0.f;
#pragma unroll
    for (int i = 0; i < 16; ++i) {
        float4 q = *(const float4*)(p + i * 4);
        s += q.x * q.x + q.y * q.y + q.z * q.z + q.w * q.w;
    }
    sq[r] = s;
}

// ---------------------------------------------------------------------------
// Kernel 3: flash-style L2-distance attention per (b,h).
// Block = 4 waves, each owns a 16-row Q tile. 16x64 K/V tiles are DOUBLE-
// BUFFERED in LDS: async copies for tile j+1 issue before the compute for
// tile j, hidden behind 32 WMMAs + softmax; s_wait_asynccnt + barrier only
// at iteration end. Stride-66 LDS rows: conflict-free, 8B aligned.
// Online softmax per lane covers 8 rows (C-layout M = v + 8*hi); row
// reductions via half-wave shfl_xor; P converts C->A layout via padded LDS.
// Grid: 96 (b,h) x 16 blocks, 128 threads.
// ---------------------------------------------------------------------------
__global__ void __launch_bounds__(128)
attn_kernel(const float* __restrict__ Qbuf, const float* __restrict__ Vbuf,
            const float* __restrict__ sq, float* __restrict__ Obuf) {
    __shared__ float Klds[2][16 * 66];
    __shared__ float Vlds[2][16 * 66];
    __shared__ float Plds[4][16 * 17];

    const int lane = threadIdx.x & 31;
    const int warp = threadIdx.x >> 5;
    const int lm = lane & 15;
    const int hi = lane >> 4;
    const int bh    = blockIdx.x >> 4;                   // 0..95
    const int itile = (blockIdx.x & 15) * 4 + warp;      // 0..63
    const int n0    = itile * 16;
    const float scale = 0.125f;                          // D^-0.5

    const float* Qb  = Qbuf + (size_t)bh * N_ * D_;
    const float* Vb  = Vbuf + (size_t)bh * N_ * D_;
    const float* sqb = sq + (size_t)bh * N_;

    // Issue the block's cooperative async copies of K/V tile j into buf.
    auto stage = [&](int j, int buf) {
        const float* ksrc = Qb + (size_t)(j * 16) * D_;
        const float* vsrc = Vb + (size_t)(j * 16) * D_;
#pragma unroll
        for (int q = 0; q < 4; ++q) {
            const int idx = threadIdx.x + q * 128;       // 512 float2 chunks
            const int row = idx >> 5, c2 = idx & 31;
            async_copy_b64(lds_off(&Klds[buf][row * 66 + c2 * 2]),
                           ksrc + row * 64 + c2 * 2);
            async_copy_b64(lds_off(&Vlds[buf][row * 66 + c2 * 2]),
                           vsrc + row * 64 + c2 * 2);
        }
    };

    // Preload all 16 A-fragments of this wave's Q row tile (16x64).
    v2f aq[16];
    {
        const float* qrow = Qb + (size_t)(n0 + lm) * D_ + 2 * hi;
#pragma unroll
        for (int s = 0; s < 16; ++s) aq[s] = *(const v2f*)(qrow + s * 4);
    }
    float snorm[8];
#pragma unroll
    for (int v = 0; v < 8; ++v) snorm[v] = sqb[n0 + v + 8 * hi];

    float m_i[8], l_i[8];
#pragma unroll
    for (int v = 0; v < 8; ++v) { m_i[v] = -1e30f; l_i[v] = 0.f; }
    v8f accO[4] = {};

    stage(0, 0);
    wait_async0();
    __syncthreads();

    for (int j = 0; j < 64; ++j) {
        const int cur = j & 1;
        if (j + 1 < 64) stage(j + 1, cur ^ 1);   // overlap with compute below
        const float* Kc = Klds[cur];
        const float* Vc = Vlds[cur];

        // S tile: 16 WMMAs over D=64.
        v8f acc = {};
#pragma unroll
        for (int s = 0; s < 16; ++s) {
            v2f b = *(const v2f*)(&Kc[lm * 66 + s * 4 + 2 * hi]);
            acc = wmma_f32(aq[s], b, acc);
        }

        const float sqm = sqb[j * 16 + lm];   // col norm for this lane's key

        float p[8], alpha[8];
#pragma unroll
        for (int v = 0; v < 8; ++v) {
            float d2 = snorm[v] + sqm - 2.0f * acc[v];
            d2 = fmaxf(d2, 0.f);
            const float sval = -scale * sqrtf(d2);
            float rmax = sval;
            rmax = fmaxf(rmax, __shfl_xor(rmax, 1));
            rmax = fmaxf(rmax, __shfl_xor(rmax, 2));
            rmax = fmaxf(rmax, __shfl_xor(rmax, 4));
            rmax = fmaxf(rmax, __shfl_xor(rmax, 8));
            const float mnew = fmaxf(m_i[v], rmax);
            p[v] = __expf(sval - mnew);
            float rsum = p[v];
            rsum += __shfl_xor(rsum, 1);
            rsum += __shfl_xor(rsum, 2);
            rsum += __shfl_xor(rsum, 4);
            rsum += __shfl_xor(rsum, 8);
            alpha[v] = __expf(m_i[v] - mnew);
            l_i[v] = l_i[v] * alpha[v] + rsum;
            m_i[v] = mnew;
        }
#pragma unroll
        for (int t = 0; t < 4; ++t)
#pragma unroll
            for (int v = 0; v < 8; ++v) accO[t][v] *= alpha[v];

        // C-layout -> A-layout via this wave's padded LDS P tile.
        float* P = Plds[warp];
#pragma unroll
        for (int v = 0; v < 8; ++v) P[(v + 8 * hi) * 17 + lm] = p[v];
        __syncthreads();   // cross-lane LDS dependency fence

        // O += P(16x16) @ V(16x64): 4 K-substeps x 4 N-tiles.
#pragma unroll
        for (int kk = 0; kk < 4; ++kk) {
            v2f aP;
            aP.x = P[lm * 17 + kk * 4 + 2 * hi];
            aP.y = P[lm * 17 + kk * 4 + 2 * hi + 1];
#pragma unroll
            for (int t = 0; t < 4; ++t) {
                v2f bV;
                bV.x = Vc[(kk * 4 + 2 * hi) * 66 + t * 16 + lm];
                bV.y = Vc[(kk * 4 + 2 * hi + 1) * 66 + t * 16 + lm];
                accO[t] = wmma_f32(aP, bV, accO[t]);
            }
        }

        wait_async0();     // next tile's copies (this wave) complete
        __syncthreads();   // all waves done copying AND done reading cur
    }

    float rl[8];
#pragma unroll
    for (int v = 0; v < 8; ++v) rl[v] = 1.0f / l_i[v];
#pragma unroll
    for (int t = 0; t < 4; ++t)
#pragma unroll
        for (int v = 0; v < 8; ++v) {
            const int r = n0 + v + 8 * hi;
            Obuf[((size_t)bh * N_ + r) * D_ + t * 16 + lm] = accO[t][v] * rl[v];
        }
}

// ---------------------------------------------------------------------------
// Kernel 4: out = attn_out([B,N,E] view of [B,H,N,D]) @ W_out^T + b_out.
// 64x64 wave tile, ping-pong K unroll; A gathered across heads (k = h*64+d;
// an even-aligned float2 never crosses a head boundary).
// Grid: 128 M-tiles x 12 N-groups = 1536 waves = 384 blocks.
// ---------------------------------------------------------------------------
__global__ void __launch_bounds__(128)
out_gemm_kernel(const float* __restrict__ Obuf, const float* __restrict__ Wout,
                const float* __restrict__ bout, float* __restrict__ out) {
    const int lane = threadIdx.x & 31;
    const int warp = threadIdx.x >> 5;
    const int wid  = blockIdx.x * 4 + warp;       // 0..1535
    const int mtile = wid % 128;
    const int ngrp  = wid / 128;                  // 0..11
    const int lm = lane & 15;
    const int hi = lane >> 4;

    const float* obase[4];
#pragma unroll
    for (int m = 0; m < 4; ++m) {
        const int arow = mtile * 64 + m * 16 + lm;      // 0..8191
        const int bb = arow >> 10, n = arow & 1023;
        obase[m] = Obuf + ((size_t)bb * H_ * N_ + n) * D_;
    }
    const float* wrow[4];
#pragma unroll
    for (int t = 0; t < 4; ++t)
        wrow[t] = Wout + (size_t)(ngrp * 64 + t * 16 + lm) * E_ + 2 * hi;

    // A-fragment gather: k = h*64 + d within [B,H,N,D].
    auto loadA = [&](int m, int k) -> v2f {
        return *(const v2f*)(obase[m] + (size_t)(k >> 6) * N_ * D_ + (k & 63));
    };

    v8f acc[4][4] = {};
    v2f a0[4], b0[4], a1[4], b1[4];
#pragma unroll
    for (int m = 0; m < 4; ++m) a0[m] = loadA(m, 2 * hi);
#pragma unroll
    for (int t = 0; t < 4; ++t) b0[t] = *(const v2f*)(wrow[t]);

    for (int k0 = 0; k0 < E_ - 8; k0 += 8) {
#pragma unroll
        for (int m = 0; m < 4; ++m) a1[m] = loadA(m, k0 + 4 + 2 * hi);
#pragma unroll
        for (int t = 0; t < 4; ++t) b1[t] = *(const v2f*)(wrow[t] + k0 + 4);
        mma16(acc, a0, b0);
#pragma unroll
        for (int m = 0; m < 4; ++m) a0[m] = loadA(m, k0 + 8 + 2 * hi);
#pragma unroll
        for (int t = 0; t < 4; ++t) b0[t] = *(const v2f*)(wrow[t] + k0 + 8);
        mma16(acc, a1, b1);
    }
    {
#pragma unroll
        for (int m = 0; m < 4; ++m) a1[m] = loadA(m, E_ - 4 + 2 * hi);
#pragma unroll
        for (int t = 0; t < 4; ++t) b1[t] = *(const v2f*)(wrow[t] + E_ - 4);
        mma16(acc, a0, b0);
        mma16(acc, a1, b1);
    }

#pragma unroll
    for (int m = 0; m < 4; ++m) {
#pragma unroll
        for (int t = 0; t < 4; ++t) {
            const int c = ngrp * 64 + t * 16 + lm;
            const float bias = bout[c];
#pragma unroll
            for (int v = 0; v < 8; ++v) {
                const int r = mtile * 64 + m * 16 + v + 8 * hi;
                out[(size_t)r * E_ + c] = acc[m][t][v] + bias;
            }
        }
    }
}

// ---------------------------------------------------------------------------
extern "C" void kernel_launch(void* const* d_in, const int* in_sizes, int n_in,
                              void* d_out, int out_size, void* d_ws, size_t ws_size,
                              hipStream_t stream) {
    (void)in_sizes; (void)n_in; (void)out_size; (void)ws_size;
    const float* x    = (const float*)d_in[0];
    const float* Wqkv = (const float*)d_in[1];
    const float* Wout = (const float*)d_in[2];
    const float* bout = (const float*)d_in[3];
    float* out = (float*)d_out;

    const size_t perTensor = (size_t)B_ * H_ * N_ * D_;   // 6,291,456 floats
    float* Qbuf = (float*)d_ws;
    float* Vbuf = Qbuf + perTensor;
    float* Obuf = Vbuf + perTensor;
    float* sqb  = Obuf + perTensor;                       // + 98,304 floats

    qkv_gemm_kernel<<<768, 128, 0, stream>>>(x, Wqkv, Qbuf, Vbuf);
    sqnorm_kernel<<<384, 256, 0, stream>>>(Qbuf, sqb);
    attn_kernel<<<1536, 128, 0, stream>>>(Qbuf, Vbuf, sqb, Obuf);
    out_gemm_kernel<<<384, 128, 0, stream>>>(Obuf, Wout, bout, out);
}